// LapLoss_103079215548
// MI455X (gfx1250) — compile-verified
//
#include <hip/hip_runtime.h>

typedef float v2f __attribute__((ext_vector_type(2)));
typedef float v8f __attribute__((ext_vector_type(8)));

// float32 of scipy gaussian_filter(sigma=1, truncate=4) 9-tap weights
#define GW0 1.3383062461474175e-4f
#define GW1 4.4318616200314320e-3f
#define GW2 5.3990966513188060e-2f
#define GW3 2.4197072451914337e-1f
#define GW4 3.9894228040143270e-1f

__device__ __forceinline__ float gw_k(int k) {
  // only called with compile-time k (fully unrolled loops) -> folds to immediates
  const float w[9] = {GW0, GW1, GW2, GW3, GW4, GW3, GW2, GW1, GW0};
  return w[k];
}

__device__ __forceinline__ float gw_sel(int d) {
  // branchless band-matrix weight: w[d] if 0<=d<=8 else 0 (d is runtime)
  float v = 0.0f;
  v = (d == 0 || d == 8) ? GW0 : v;
  v = (d == 1 || d == 7) ? GW1 : v;
  v = (d == 2 || d == 6) ? GW2 : v;
  v = (d == 3 || d == 5) ? GW3 : v;
  v = (d == 4)           ? GW4 : v;
  return v;
}

// ---------------- elementwise diff ----------------
__global__ __launch_bounds__(256) void k_diff(const float* __restrict__ a,
                                              const float* __restrict__ b,
                                              float* __restrict__ o, int n) {
  int i = blockIdx.x * 256 + threadIdx.x;
  if (i < n) o[i] = a[i] - b[i];
}

// ------- generic 9-tap reflect Gaussian along a strided axis (B/D/H) -------
__global__ __launch_bounds__(256) void k_gauss_axis(const float* __restrict__ in,
                                                    float* __restrict__ out,
                                                    int n, int stride, int total) {
  int i = blockIdx.x * 256 + threadIdx.x;
  if (i >= total) return;
  int pos  = (i / stride) % n;
  int base = i - pos * stride;
  int m2 = 2 * n;
  float acc = 0.0f;
#pragma unroll
  for (int k = 0; k < 9; ++k) {
    int p = pos + k - 4;
    int q = p % m2;
    if (q < 0) q += m2;          // handles tiny axes (n=2, n=4) with multi-wrap reflect
    if (q >= n) q = m2 - 1 - q;
    acc += gw_k(k) * in[base + q * stride];
  }
  out[i] = acc;
}

// ------- W-axis 9-tap conv as banded matmul: D(16x16) = A(16x24) x B(24x16) -------
// 6 chained V_WMMA_F32_16X16X4_F32 per wave; full f32 precision.
__global__ __launch_bounds__(128) void k_gauss_w_wmma(const float* __restrict__ in,
                                                      float* __restrict__ out,
                                                      int W, int R) {
  int lane = threadIdx.x & 31;
  int wave = threadIdx.x >> 5;
  int tile = blockIdx.x * 4 + wave;        // 4 waves per block, grids are exact
  int tilesW = W >> 4;
  int tilesR = R >> 4;
  if (tile >= tilesW * tilesR) return;     // wave-uniform; EXEC stays all-1s
  int tw = tile % tilesW;
  int tr = tile / tilesW;
  int w0 = tw << 4;
  int rowBase = (tr << 4) * W;
  int half = lane >> 4;                    // 0: K even-pair base 0, 1: base 2
  int ln   = lane & 15;                    // M for A-frag, N for B/C/D frags

  v8f acc = {0.f, 0.f, 0.f, 0.f, 0.f, 0.f, 0.f, 0.f};
#pragma unroll
  for (int s = 0; s < 6; ++s) {
    v2f a, bm;
#pragma unroll
    for (int j = 0; j < 2; ++j) {
      int p  = s * 4 + half * 2 + j;       // 0..23 along the 24-wide input window
      int wi = w0 - 4 + p;                 // reflect (single bounce valid: W >= 16)
      if (wi < 0) wi = -wi - 1;
      else if (wi >= W) wi = 2 * W - 1 - wi;
      a[j]  = in[rowBase + ln * W + wi];   // A[M=ln][K=p]
      bm[j] = gw_sel(p - ln);              // B[K=p][N=ln] = w[p-n] banded
    }
    acc = __builtin_amdgcn_wmma_f32_16x16x4_f32(
        /*neg_a=*/false, a, /*neg_b=*/false, bm,
        /*c_mod=*/(short)0, acc, /*reuse_a=*/false, /*reuse_b=*/false);
  }
  // C/D layout: lane<16 -> N=ln, M=r ; lane>=16 -> N=ln, M=8+r
#pragma unroll
  for (int r = 0; r < 8; ++r) {
    out[rowBase + (half * 8 + r) * W + (w0 + ln)] = acc[r];
  }
}

// ---------------- stride-2 downsample on D,H,W ----------------
__global__ __launch_bounds__(256) void k_down(const float* __restrict__ in,
                                              float* __restrict__ out,
                                              int D, int H, int W, int total_out) {
  int i = blockIdx.x * 256 + threadIdx.x;
  if (i >= total_out) return;
  int W2 = W >> 1, H2 = H >> 1, D2 = D >> 1;
  int w = i % W2; int t = i / W2;
  int h = t % H2; t /= H2;
  int d = t % D2; int b = t / D2;
  out[i] = in[((b * D + 2 * d) * H + 2 * h) * W + 2 * w];
}

// ------- zero-stuff upsample (D,H interleave; zeros APPENDED along W) x8 -------
__global__ __launch_bounds__(256) void k_stuff(const float* __restrict__ dn,
                                               float* __restrict__ out,
                                               int D, int H, int W, int total) {
  int i = blockIdx.x * 256 + threadIdx.x;
  if (i >= total) return;
  int w = i % W; int t = i / W;
  int h = t % H; t /= H;
  int d = t % D; int b = t / D;
  int W2 = W >> 1, H2 = H >> 1, D2 = D >> 1;
  float v = 0.0f;
  if (((d | h) & 1) == 0 && w < W2)
    v = 8.0f * dn[((b * D2 + (d >> 1)) * H2 + (h >> 1)) * W2 + w];
  out[i] = v;
}

// ---------------- deterministic two-stage |a-b| reduction ----------------
__global__ __launch_bounds__(256) void k_absdiff_reduce(const float* __restrict__ a,
                                                        const float* __restrict__ b,
                                                        int n, float* __restrict__ partials) {
  float s = 0.0f;
  for (int i = blockIdx.x * 256 + threadIdx.x; i < n; i += gridDim.x * 256)
    s += fabsf(a[i] - b[i]);
  __shared__ float sh[256];
  sh[threadIdx.x] = s;
  __syncthreads();
#pragma unroll
  for (int off = 128; off > 0; off >>= 1) {
    if (threadIdx.x < off) sh[threadIdx.x] += sh[threadIdx.x + off];
    __syncthreads();
  }
  if (threadIdx.x == 0) partials[blockIdx.x] = sh[0];
}

__global__ __launch_bounds__(256) void k_sum_partials(const float* __restrict__ partials,
                                                      int n, float* __restrict__ acc, int slot) {
  float s = 0.0f;
  for (int i = threadIdx.x; i < n; i += 256) s += partials[i];
  __shared__ float sh[256];
  sh[threadIdx.x] = s;
  __syncthreads();
#pragma unroll
  for (int off = 128; off > 0; off >>= 1) {
    if (threadIdx.x < off) sh[threadIdx.x] += sh[threadIdx.x + off];
    __syncthreads();
  }
  if (threadIdx.x == 0) acc[slot] = sh[0];
}

__global__ void k_finalize(const float* __restrict__ acc, float* __restrict__ out) {
  out[0] = acc[0] * (1.0f / 8388608.0f)
         + acc[1] * (1.0f / 1048576.0f)
         + acc[2] * (1.0f / 131072.0f)
         + acc[3] * (1.0f / 16384.0f)
         + acc[4] * (1.0f / 2048.0f);
}

extern "C" void kernel_launch(void* const* d_in, const int* in_sizes, int n_in,
                              void* d_out, int out_size, void* d_ws, size_t ws_size,
                              hipStream_t stream) {
  const float* input  = (const float*)d_in[0];
  const float* target = (const float*)d_in[1];
  float* out = (float*)d_out;
  float* ws  = (float*)d_ws;

  const int N0 = 2 * 1 * 64 * 256 * 256;         // 8,388,608
  float* cur      = ws;                          // pyramid "cur" (level 0 = diff)
  float* X        = ws + (size_t)N0;
  float* Y        = ws + (size_t)2 * N0;
  float* downBuf0 = ws + (size_t)3 * N0;         // N0/8
  float* downBuf1 = downBuf0 + N0 / 8;           // N0/8 (oversized for deep levels)
  float* partials = downBuf1 + N0 / 8;           // 1024 floats
  float* acc      = partials + 1024;             // 5 floats

  // Pyramid of (input - target): linearity => lap_in - lap_tg == lap(diff).
  k_diff<<<N0 / 256, 256, 0, stream>>>(input, target, cur, N0);

  int D = 64, H = 256, W = 256;
  for (int l = 0; l < 5; ++l) {
    int total = 2 * D * H * W;
    int strideB = D * H * W, strideD = H * W, strideH = W;
    int R = 2 * D * H;
    int tiles = (R >> 4) * (W >> 4);             // divisible by 4 at every level
    int gb = total / 256;

    // filt = G(cur): axes B (n=2), D, H via scalar; W via WMMA. (C axis is identity.)
    k_gauss_axis<<<gb, 256, 0, stream>>>(cur, X, 2, strideB, total);
    k_gauss_axis<<<gb, 256, 0, stream>>>(X, Y, D, strideD, total);
    k_gauss_axis<<<gb, 256, 0, stream>>>(Y, X, H, strideH, total);
    k_gauss_w_wmma<<<tiles / 4, 128, 0, stream>>>(X, Y, W, R);   // filt -> Y

    // down = filt[::2,::2,::2]
    int D2 = D >> 1, H2 = H >> 1, W2 = W >> 1;
    int totalD = 2 * D2 * H2 * W2;
    float* dn = (l & 1) ? downBuf1 : downBuf0;
    k_down<<<(totalD + 255) / 256, 256, 0, stream>>>(Y, dn, D, H, W, totalD);

    // up = G(8 * zero_stuff(down))
    k_stuff<<<gb, 256, 0, stream>>>(dn, X, D, H, W, total);
    k_gauss_axis<<<gb, 256, 0, stream>>>(X, Y, 2, strideB, total);
    k_gauss_axis<<<gb, 256, 0, stream>>>(Y, X, D, strideD, total);
    k_gauss_axis<<<gb, 256, 0, stream>>>(X, Y, H, strideH, total);
    k_gauss_w_wmma<<<tiles / 4, 128, 0, stream>>>(Y, X, W, R);   // up -> X

    // loss_l = mean(|cur - up|), deterministic two-stage reduction
    k_absdiff_reduce<<<1024, 256, 0, stream>>>(cur, X, total, partials);
    k_sum_partials<<<1, 256, 0, stream>>>(partials, 1024, acc, l);

    cur = dn; D = D2; H = H2; W = W2;
  }

  k_finalize<<<1, 1, 0, stream>>>(acc, out);
}